// Hbv_2f_2929167696147
// MI455X (gfx1250) — compile-verified
//
#include <hip/hip_runtime.h>
#include <math.h>

#define NEARZERO 1e-5f
#define RLEN 15
#define PF_DIST 8   // prefetch distance in timesteps
#define LOG2E 1.4426950408889634f

// Direct hardware transcendentals (v_log_f32 = log2, v_exp_f32 = exp2)
__device__ __forceinline__ float fast_log2(float x) { return __builtin_amdgcn_logf(x); }
__device__ __forceinline__ float fast_exp2(float x) { return __builtin_amdgcn_exp2f(x); }
__device__ __forceinline__ float fast_exp(float x)  { return __builtin_amdgcn_exp2f(x * LOG2E); }

__global__ __launch_bounds__(256) void hbv_scan_kernel(
    const float* __restrict__ forcing,   // (T, G, 3): prcp, tmean, pet
    const float* __restrict__ dynr,      // (T, G, 3): beta, k0, betaet (raw)
    const float* __restrict__ sraw,      // (G, 15): 13 static + 2 routing (raw)
    float* __restrict__ out,             // (T, G): routed discharge
    int T, int G)
{
    const int g = blockIdx.x * blockDim.x + threadIdx.x;
    if (g >= G) return;

    // ---- static parameters (descale once) ----
    const float* s = sraw + (size_t)g * 15;
    const float FC    = 50.0f  + s[0]  * 950.0f;
    const float K1    = 0.01f  + s[1]  * 0.49f;
    const float K2    = 0.001f + s[2]  * 0.199f;
    const float LP    = 0.2f   + s[3]  * 0.8f;
    const float PERCp = s[4]  * 10.0f;
    const float UZL   = s[5]  * 100.0f;
    const float TT    = -2.5f  + s[6]  * 5.0f;
    const float CFMAX = 0.5f   + s[7]  * 9.5f;
    const float CFR   = s[8]  * 0.1f;
    const float CWH   = s[9]  * 0.2f;
    const float Cpar  = s[10];
    const float rout_a = s[13] * 2.9f;
    const float rout_b = s[14] * 6.5f;

    // ---- per-cell gamma routing weights (normalization cancels gamma/theta^aa) ----
    const float aa    = fmaxf(rout_a, 0.0f) + 0.1f;
    const float theta = fmaxf(rout_b, 0.0f) + 0.5f;
    const float am1   = aa - 1.0f;
    const float invth = 1.0f / theta;
    float w[RLEN];
    float wsum = 0.0f;
#pragma unroll
    for (int k = 0; k < RLEN; ++k) {
        const float tk = (float)k + 0.5f;
        const float v = fast_exp2(am1 * fast_log2(tk)) * fast_exp(-tk * invth);
        w[k] = v;
        wsum += v;
    }
    const float winv = 1.0f / wsum;
#pragma unroll
    for (int k = 0; k < RLEN; ++k) w[k] *= winv;

    // ---- ring buffer of recent Q (fused routing), held in VGPRs ----
    float q[RLEN];
#pragma unroll
    for (int k = 0; k < RLEN; ++k) q[k] = 0.0f;

    // ---- model state ----
    float SNOWPACK = 1e-3f, MELTWATER = 1e-3f, SM = 1e-3f, SUZ = 1e-3f, SLZ = 1e-3f;

    const float invFC   = 1.0f / FC;
    const float invLPFC = 1.0f / (LP * FC);

    const size_t stride = (size_t)G * 3;
    const float* fp = forcing + (size_t)g * 3;
    const float* dp = dynr    + (size_t)g * 3;

    // software pipeline: preload step 0
    float P  = fp[0], TE  = fp[1], PET  = fp[2];
    float d0 = dp[0], d1 = dp[1], d2 = dp[2];

    for (int t = 0; t < T; ++t) {
        // gfx1250 prefetch path: pull stream PF_DIST steps ahead into cache
        {
            const size_t pfo = (t + PF_DIST < T) ? stride * PF_DIST : 0;
            __builtin_prefetch(fp + pfo, 0, 1);
            __builtin_prefetch(dp + pfo, 0, 1);
        }
        // register double-buffer: issue next-step loads before the dependent ALU chain
        float Pn = 0.f, TEn = 0.f, PETn = 0.f, d0n = 0.f, d1n = 0.f, d2n = 0.f;
        if (t + 1 < T) {
            const float* fpn = fp + stride;
            const float* dpn = dp + stride;
            Pn  = fpn[0]; TEn = fpn[1]; PETn = fpn[2];
            d0n = dpn[0]; d1n = dpn[1]; d2n = dpn[2];
        }

        // dynamic params for this step
        const float BETA   = 1.0f  + d0 * 5.0f;
        const float K0     = 0.05f + d1 * 0.85f;
        const float BETAET = 0.3f  + d2 * 4.7f;

        // ---- HBV step (order exactly as reference) ----
        const float is_rain = (TE >= TT) ? 1.0f : 0.0f;
        const float RAIN = P * is_rain;
        const float SNOW = P - RAIN;
        SNOWPACK += SNOW;

        const float melt = fminf(fmaxf(CFMAX * (TE - TT), 0.0f), SNOWPACK);
        MELTWATER += melt;
        SNOWPACK  -= melt;

        const float refreeze = fminf(fmaxf(CFR * CFMAX * (TT - TE), 0.0f), MELTWATER);
        SNOWPACK  += refreeze;
        MELTWATER -= refreeze;

        const float tosoil = fmaxf(MELTWATER - CWH * SNOWPACK, 0.0f);
        MELTWATER -= tosoil;

        // soil wetness: (SM/FC)^BETA, SM > 0 guaranteed
        const float sw = fminf(fmaxf(fast_exp2(BETA * fast_log2(SM * invFC)), 0.0f), 1.0f);
        const float recharge = (RAIN + tosoil) * sw;
        SM = SM + RAIN + tosoil - recharge;

        const float excess = fmaxf(SM - FC, 0.0f);
        SM -= excess;

        const float ef = fminf(fmaxf(fast_exp2(BETAET * fast_log2(SM * invLPFC)), 0.0f), 1.0f);
        const float ETact = fminf(SM, PET * ef);
        SM = fmaxf(SM - ETact, NEARZERO);

        const float capillary = fminf(SLZ, Cpar * SLZ * (1.0f - fminf(SM * invFC, 1.0f)));
        SM  = fmaxf(SM + capillary, NEARZERO);
        SLZ = fmaxf(SLZ - capillary, NEARZERO);

        SUZ = SUZ + recharge + excess;
        const float PERC = fminf(SUZ, PERCp);
        SUZ -= PERC;

        const float Q0 = K0 * fmaxf(SUZ - UZL, 0.0f);
        SUZ -= Q0;
        const float Q1 = K1 * SUZ;
        SUZ -= Q1;

        SLZ = fmaxf(SLZ + PERC, 0.0f);
        const float Q2 = K2 * SLZ;
        SLZ -= Q2;

        const float Q = Q0 + Q1 + Q2;

        // ---- fused 15-tap gamma routing (all registers) ----
#pragma unroll
        for (int k = RLEN - 1; k > 0; --k) q[k] = q[k - 1];
        q[0] = Q;
        float acc = 0.0f;
#pragma unroll
        for (int k = 0; k < RLEN; ++k) acc = fmaf(w[k], q[k], acc);

        out[(size_t)t * G + g] = acc;

        // rotate pipeline
        fp += stride; dp += stride;
        P = Pn; TE = TEn; PET = PETn;
        d0 = d0n; d1 = d1n; d2 = d2n;
    }
}

extern "C" void kernel_launch(void* const* d_in, const int* in_sizes, int n_in,
                              void* d_out, int out_size, void* d_ws, size_t ws_size,
                              hipStream_t stream) {
    const float* forcing = (const float*)d_in[0];   // (T, G, 3)
    const float* dynr    = (const float*)d_in[1];   // (T, G, 3)
    const float* sraw    = (const float*)d_in[2];   // (G, 15)
    float* out = (float*)d_out;                     // (T, G)

    const int G = in_sizes[2] / 15;
    const int T = in_sizes[0] / (3 * G);

    const int block = 256;
    const int grid  = (G + block - 1) / block;
    hipLaunchKernelGGL(hbv_scan_kernel, dim3(grid), dim3(block), 0, stream,
                       forcing, dynr, sraw, out, T, G);
}